// GraphConvolution_1580547973936
// MI455X (gfx1250) — compile-verified
//
#include <hip/hip_runtime.h>

// ---------------------------------------------------------------------------
// GCN layer: out = segment_sum(edge_val * (x@W)[edge_col], edge_row) + bias
// N_NODES=100000, N_EDGES=1600000, D_IN=D_OUT=128
// ---------------------------------------------------------------------------

typedef __attribute__((ext_vector_type(2))) float v2f;
typedef __attribute__((ext_vector_type(8))) float v8f;

#define D_DIM 128
#define LXS 132   // padded LDS stride for x tile  (bank-conflict avoidance)
#define LWS 136   // padded LDS stride for weight  (half-wave shift = 16 banks)

// ---------------------------------------------------------------------------
// Kernel 1: support = x @ W  via V_WMMA_F32_16X16X4_F32
// grid.x = ceil(n_nodes/16); block = 256 (8 waves); wave w owns N-tile w.
// ---------------------------------------------------------------------------
__global__ __launch_bounds__(256) void gcn_gemm_wmma(
    const float* __restrict__ x, const float* __restrict__ w,
    float* __restrict__ support, int n_nodes) {
  __shared__ float lx[16 * LXS];       // 16 x 128 x-tile (padded)
  __shared__ float lw[D_DIM * LWS];    // 128 x 128 weight (padded)

  const int tid   = threadIdx.x;
  const int tileM = blockIdx.x;
  const long xbase = (long)tileM * 16 * D_DIM;
  const long xlim  = (long)n_nodes * D_DIM;

  // Stage x tile: 2048 floats, 8 per thread
  for (int i = tid; i < 16 * D_DIM; i += 256) {
    int r = i >> 7, c = i & (D_DIM - 1);
    long g = xbase + i;
    lx[r * LXS + c] = (g < xlim) ? x[g] : 0.0f;
  }
  // Stage weight: 16384 floats, 64 per thread
  for (int i = tid; i < D_DIM * D_DIM; i += 256) {
    int r = i >> 7, c = i & (D_DIM - 1);
    lw[r * LWS + c] = w[i];
  }
  __syncthreads();

  const int wave = tid >> 5;          // N-tile index 0..7
  const int lane = tid & 31;
  const int mn   = lane & 15;         // A: row M ; B/D: col N
  const int kb   = (lane >> 4) * 2;   // half-wave K-pair base (0 or 2)
  const int ncol = wave * 16 + mn;

  v8f acc = {};
#pragma unroll
  for (int k0 = 0; k0 < D_DIM; k0 += 4) {
    v2f a, b;
    // A 16x4 fragment: lane holds x[m][k0+kb], x[m][k0+kb+1]  (ds_load_b64)
    a.x = lx[mn * LXS + k0 + kb];
    a.y = lx[mn * LXS + k0 + kb + 1];
    // B 4x16 fragment: lane holds W[k0+kb][n], W[k0+kb+1][n]
    b.x = lw[(k0 + kb) * LWS + ncol];
    b.y = lw[(k0 + kb + 1) * LWS + ncol];
    acc = __builtin_amdgcn_wmma_f32_16x16x4_f32(
        /*neg_a=*/false, a, /*neg_b=*/false, b,
        /*c_mod=*/(short)0, acc, /*reuse_a=*/false, /*reuse_b=*/false);
  }

  // D 16x16: VGPR g -> row (g + 8*(lane/16)), col = lane%16
  const int rowHalf = (lane >> 4) * 8;
#pragma unroll
  for (int g = 0; g < 8; ++g) {
    long row = (long)tileM * 16 + g + rowHalf;
    if (row < n_nodes) support[row * D_DIM + ncol] = acc[g];
  }
}

// ---------------------------------------------------------------------------
// Kernel 2: out[i] = bias[i % 128]   (seed accumulator with bias)
// ---------------------------------------------------------------------------
__global__ __launch_bounds__(256) void gcn_bias_init(
    float* __restrict__ out, const float* __restrict__ bias, long total) {
  long i = (long)blockIdx.x * blockDim.x + threadIdx.x;
  if (i < total) out[i] = bias[i & (D_DIM - 1)];
}

// ---------------------------------------------------------------------------
// Kernel 3: scatter-add messages. One wave per edge; lane owns a float4
// chunk. Native L2 float atomics via inline asm (global_atomic_add_f32).
// ---------------------------------------------------------------------------
__device__ __forceinline__ void atomic_add_f32_global(float* p, float v) {
  asm volatile("global_atomic_add_f32 %0, %1, off"
               :
               : "v"(p), "v"(v)
               : "memory");
}

__global__ __launch_bounds__(256) void gcn_edge_scatter(
    const int* __restrict__ edge_row, const int* __restrict__ edge_col,
    const float* __restrict__ edge_val, const float* __restrict__ support,
    float* __restrict__ out, int n_edges) {
  long t = (long)blockIdx.x * blockDim.x + threadIdx.x;
  int e = (int)(t >> 5);
  if (e >= n_edges) return;
  int lane = (int)(t & 31);

  int r = edge_row[e];
  int c = edge_col[e];
  float v = edge_val[e];

  const float4* src =
      (const float4*)(support + (long)c * D_DIM) + lane;  // global_load_b128
  float4 s = *src;

  float* dst = out + (long)r * D_DIM + lane * 4;
  atomic_add_f32_global(dst + 0, v * s.x);
  atomic_add_f32_global(dst + 1, v * s.y);
  atomic_add_f32_global(dst + 2, v * s.z);
  atomic_add_f32_global(dst + 3, v * s.w);
}

// ---------------------------------------------------------------------------
extern "C" void kernel_launch(void* const* d_in, const int* in_sizes, int n_in,
                              void* d_out, int out_size, void* d_ws,
                              size_t ws_size, hipStream_t stream) {
  const float* x        = (const float*)d_in[0];
  const int*   edge_row = (const int*)d_in[1];
  const int*   edge_col = (const int*)d_in[2];
  const float* edge_val = (const float*)d_in[3];
  const float* weight   = (const float*)d_in[4];
  const float* bias     = (const float*)d_in[5];
  float*       out      = (float*)d_out;
  float*       support  = (float*)d_ws;   // n_nodes * 128 floats (51.2 MB)

  const int n_nodes = in_sizes[0] / D_DIM;
  const int n_edges = in_sizes[1];

  // 1) support = x @ W  (WMMA f32)
  int mtiles = (n_nodes + 15) / 16;
  gcn_gemm_wmma<<<mtiles, 256, 0, stream>>>(x, weight, support, n_nodes);

  // 2) out = broadcast(bias)
  long total = (long)n_nodes * D_DIM;
  int bblocks = (int)((total + 255) / 256);
  gcn_bias_init<<<bblocks, 256, 0, stream>>>(out, bias, total);

  // 3) out += scatter-add of edge messages
  long sthreads = (long)n_edges * 32;
  int sblocks = (int)((sthreads + 255) / 256);
  gcn_edge_scatter<<<sblocks, 256, 0, stream>>>(edge_row, edge_col, edge_val,
                                                support, out, n_edges);
}